// SelfConditioningResidualLayer_28192165331649
// MI455X (gfx1250) — compile-verified
//
#include <hip/hip_runtime.h>
#include <math.h>

typedef __attribute__((ext_vector_type(2))) float v2f;
typedef __attribute__((ext_vector_type(8))) float v8f;

#define D_NODE     128
#define RBF_DIM    16
#define N_A        16
#define DIN_N      160        // 16 rbf + 16 a + 128 s_t
#define NIN_STRIDE 164        // padded: conflict-free frag reads, 16B-aligned rows
#define H_STRIDE   132        // padded: 132 % 64 = 4 -> conflict-free
#define D_EDGE     64
#define N_E        5
#define DIN_E      85
#define K_EDGE     92         // 85 padded to multiple of 4; stride 92 (16B-aligned rows)

__device__ __forceinline__ float silu_f(float x) {
    // x * sigmoid(x), sigmoid via fast hardware reciprocal (v_rcp_f32)
    return x * __builtin_amdgcn_rcpf(1.0f + __expf(-x));
}

__device__ __forceinline__ v8f wmma_f32(v2f a, v2f b, v8f c) {
    // D = A(16x4 f32) * B(4x16 f32) + C(16x16 f32)
    return __builtin_amdgcn_wmma_f32_16x16x4_f32(
        /*neg_a=*/false, a, /*neg_b=*/false, b,
        /*c_mod=*/(short)0, c, /*reuse_a=*/false, /*reuse_b=*/false);
}

// Async 16B global -> LDS copy (ASYNCcnt-tracked, no VGPR data path).
// lds_b: low 32 bits of generic shared pointer == LDS byte offset.
__device__ __forceinline__ void async_g2l_b128(unsigned lds_b, const void* gptr) {
    unsigned long long ga = (unsigned long long)gptr;
    asm volatile("global_load_async_to_lds_b128 %0, %1, off"
                 :: "v"(lds_b), "v"(ga) : "memory");
}
__device__ __forceinline__ void wait_async0() {
    asm volatile("s_wait_asynccnt 0x0" ::: "memory");
}
__device__ __forceinline__ unsigned lds_off(const void* p) {
    return (unsigned)(unsigned long long)p;
}

// ---------------------------------------------------------------------------
// Node path: features -> GEMM1(160x128) -> silu -> GEMM2(128x128) -> silu -> +s_t
// 128 rows per block, 8 wave32s, each wave owns a 16-row x 128-col stripe.
// ---------------------------------------------------------------------------
__global__ __launch_bounds__(256)
void node_mlp_kernel(const float* __restrict__ s_t,
                     const float* __restrict__ x_t,
                     const float* __restrict__ v_t,
                     const float* __restrict__ dst_lig_x,
                     const float* __restrict__ dst_lig_a,
                     const float* __restrict__ W1, const float* __restrict__ b1,
                     const float* __restrict__ W2, const float* __restrict__ b2,
                     float* __restrict__ node_out,
                     float* __restrict__ v_out,
                     int N)
{
    __shared__ __align__(16) float nin[128 * NIN_STRIDE];  // concat(rbf, a, s_t)
    __shared__ __align__(16) float hln[128 * H_STRIDE];    // hidden activations

    const int tid     = threadIdx.x;
    const int rowBase = blockIdx.x * 128;
    const bool fullTile = (rowBase + 128) <= N;

    // ---- s_t tile into nin[:, 32:160] --------------------------------------
    if (fullTile) {
        // async DMA: 128 rows x 32 16B-chunks = 4096 chunks, 16 per thread
#pragma unroll 4
        for (int i = 0; i < 16; ++i) {
            const int idx = tid + 256 * i;
            const int r = idx >> 5, cg = idx & 31;
            async_g2l_b128(lds_off(&nin[r * NIN_STRIDE + 32 + cg * 4]),
                           &s_t[(size_t)(rowBase + r) * D_NODE + cg * 4]);
        }
    } else {
#pragma unroll 4
        for (int i = 0; i < 64; ++i) {
            const int idx = tid + 256 * i;
            const int r = idx >> 7, c = idx & 127;
            const int gr = rowBase + r;
            nin[r * NIN_STRIDE + 32 + c] = (gr < N) ? s_t[(size_t)gr * D_NODE + c] : 0.0f;
        }
    }

    // ---- feature stage -----------------------------------------------------
    if (tid < 128) {
        const int r = rowBase + tid;
        float* row = &nin[tid * NIN_STRIDE];
        if (r < N) {
            const float dx = dst_lig_x[r * 3 + 0] - x_t[r * 3 + 0];
            const float dy = dst_lig_x[r * 3 + 1] - x_t[r * 3 + 1];
            const float dz = dst_lig_x[r * 3 + 2] - x_t[r * 3 + 2];
            float d2 = dx * dx + dy * dy + dz * dz;
            d2 = fmaxf(d2, 1e-8f);
            const float d = sqrtf(d2);
            const float inv_sigma = 16.0f / 10.0f;      // 1/sigma
            const float mu_step   = 10.0f / 15.0f;      // linspace(0,10,16) step
#pragma unroll
            for (int j = 0; j < RBF_DIM; ++j) {
                const float t = (d - mu_step * (float)j) * inv_sigma;
                row[j] = __expf(-t * t);
            }
#pragma unroll
            for (int j = 0; j < N_A; ++j)
                row[RBF_DIM + j] = dst_lig_a[r * N_A + j];

            // v_out: copy v_t, overwrite last row with x_diff / dij
            const float invd = __builtin_amdgcn_rcpf(d);
#pragma unroll
            for (int j = 0; j < 9; ++j)
                v_out[r * 12 + j] = v_t[r * 12 + j];
            v_out[r * 12 + 9]  = dx * invd;
            v_out[r * 12 + 10] = dy * invd;
            v_out[r * 12 + 11] = dz * invd;
        } else {
#pragma unroll
            for (int j = 0; j < RBF_DIM + N_A; ++j) row[j] = 0.0f;
        }
    }
    wait_async0();
    __syncthreads();

    const int lane = tid & 31;
    const int wv   = tid >> 5;
    const int lo   = lane & 15;
    const int hi   = lane >> 4;

    // ---- GEMM1: nin(128x160) @ W1(160x128) + b1 ----------------------------
    v8f acc[8];
#pragma unroll
    for (int nb = 0; nb < 8; ++nb) {
        const float bv = b1[nb * 16 + lo];
#pragma unroll
        for (int i = 0; i < 8; ++i) acc[nb][i] = bv;
    }
    {
        const float* arow = &nin[(wv * 16 + lo) * NIN_STRIDE + hi * 2];
#pragma unroll 2
        for (int kb = 0; kb < DIN_N; kb += 4) {
            v2f a;
            a.x = arow[kb];
            a.y = arow[kb + 1];
            const int kidx = (kb + hi * 2) * D_NODE + lo;
#pragma unroll
            for (int nb = 0; nb < 8; ++nb) {
                v2f b;
                b.x = W1[kidx + nb * 16];
                b.y = W1[kidx + D_NODE + nb * 16];
                acc[nb] = wmma_f32(a, b, acc[nb]);
            }
        }
    }
    // silu -> stage h in LDS
#pragma unroll
    for (int nb = 0; nb < 8; ++nb) {
#pragma unroll
        for (int i = 0; i < 8; ++i) {
            const int m = i + hi * 8;
            hln[(wv * 16 + m) * H_STRIDE + nb * 16 + lo] = silu_f(acc[nb][i]);
        }
    }
    __syncthreads();

    // ---- GEMM2: h(128x128) @ W2(128x128) + b2 ------------------------------
    v8f acc2[8];
#pragma unroll
    for (int nb = 0; nb < 8; ++nb) {
        const float bv = b2[nb * 16 + lo];
#pragma unroll
        for (int i = 0; i < 8; ++i) acc2[nb][i] = bv;
    }
    {
        const float* arow = &hln[(wv * 16 + lo) * H_STRIDE + hi * 2];
#pragma unroll 2
        for (int kb = 0; kb < D_NODE; kb += 4) {
            v2f a;
            a.x = arow[kb];
            a.y = arow[kb + 1];
            const int kidx = (kb + hi * 2) * D_NODE + lo;
#pragma unroll
            for (int nb = 0; nb < 8; ++nb) {
                v2f b;
                b.x = W2[kidx + nb * 16];
                b.y = W2[kidx + D_NODE + nb * 16];
                acc2[nb] = wmma_f32(a, b, acc2[nb]);
            }
        }
    }
    // ---- epilogue: node_out = s_t + silu(.) --------------------------------
#pragma unroll
    for (int nb = 0; nb < 8; ++nb) {
#pragma unroll
        for (int i = 0; i < 8; ++i) {
            const int m  = i + hi * 8;
            const int gr = rowBase + wv * 16 + m;
            if (gr < N) {
                const int c = nb * 16 + lo;
                const float sv = nin[(wv * 16 + m) * NIN_STRIDE + 32 + c];
                node_out[(size_t)gr * D_NODE + c] = sv + silu_f(acc2[nb][i]);
            }
        }
    }
}

// ---------------------------------------------------------------------------
// Edge path: gather + rbf-diff features -> GEMM(85x64) -> silu -> +e_u, dup.
// 128 edges per block, 8 wave32s, each wave owns a 16-row x 64-col stripe.
// ---------------------------------------------------------------------------
__global__ __launch_bounds__(256)
void edge_mlp_kernel(const float* __restrict__ e_t,
                     const float* __restrict__ x_t,
                     const float* __restrict__ dst_lig_x,
                     const float* __restrict__ dst_lig_e,
                     const long long* __restrict__ edge_src,
                     const long long* __restrict__ edge_dst,
                     const float* __restrict__ We, const float* __restrict__ be,
                     float* __restrict__ e_out,
                     int Eu)
{
    __shared__ __align__(16) float ein[128 * K_EDGE];  // concat(e_u, lig_e, d_input)+pad

    const int tid   = threadIdx.x;
    const int eBase = blockIdx.x * 128;
    const bool fullTile = (eBase + 128) <= Eu;

    // ---- e_u tile into ein[:, 0:64] ----------------------------------------
    if (fullTile) {
        // async DMA: 128 rows x 16 16B-chunks = 2048 chunks, 8 per thread
#pragma unroll 4
        for (int i = 0; i < 8; ++i) {
            const int idx = tid + 256 * i;
            const int r = idx >> 4, cg = idx & 15;
            async_g2l_b128(lds_off(&ein[r * K_EDGE + cg * 4]),
                           &e_t[(size_t)(eBase + r) * D_EDGE + cg * 4]);
        }
    } else {
#pragma unroll 4
        for (int i = 0; i < 32; ++i) {
            const int idx = tid + 256 * i;
            const int r = idx >> 6, c = idx & 63;
            const int ge = eBase + r;
            ein[r * K_EDGE + c] = (ge < Eu) ? e_t[(size_t)ge * D_EDGE + c] : 0.0f;
        }
    }

    if (tid < 128) {
        const int e = eBase + tid;
        float* row = &ein[tid * K_EDGE];
        if (e < Eu) {
            const long long su = edge_src[e];
            const long long du = edge_dst[e];
            float ax = x_t[su * 3 + 0] - x_t[du * 3 + 0];
            float ay = x_t[su * 3 + 1] - x_t[du * 3 + 1];
            float az = x_t[su * 3 + 2] - x_t[du * 3 + 2];
            const float d_t = sqrtf(fmaxf(ax * ax + ay * ay + az * az, 1e-8f));
            ax = dst_lig_x[su * 3 + 0] - dst_lig_x[du * 3 + 0];
            ay = dst_lig_x[su * 3 + 1] - dst_lig_x[du * 3 + 1];
            az = dst_lig_x[su * 3 + 2] - dst_lig_x[du * 3 + 2];
            const float d_1 = sqrtf(fmaxf(ax * ax + ay * ay + az * az, 1e-8f));
            const float inv_sigma = 16.0f / 10.0f;
            const float mu_step   = 10.0f / 15.0f;
#pragma unroll
            for (int j = 0; j < RBF_DIM; ++j) {
                const float mu = mu_step * (float)j;
                const float t1 = (d_1 - mu) * inv_sigma;
                const float t0 = (d_t - mu) * inv_sigma;
                row[D_EDGE + N_E + j] = __expf(-t1 * t1) - __expf(-t0 * t0);
            }
#pragma unroll
            for (int j = 0; j < N_E; ++j)
                row[D_EDGE + j] = dst_lig_e[(size_t)e * N_E + j];
#pragma unroll
            for (int j = DIN_E; j < K_EDGE; ++j) row[j] = 0.0f;   // K padding
        } else {
#pragma unroll
            for (int j = D_EDGE; j < K_EDGE; ++j) row[j] = 0.0f;
        }
    }
    wait_async0();
    __syncthreads();

    const int lane = tid & 31;
    const int wv   = tid >> 5;
    const int lo   = lane & 15;
    const int hi   = lane >> 4;

    v8f acc[4];
#pragma unroll
    for (int nb = 0; nb < 4; ++nb) {
        const float bv = be[nb * 16 + lo];
#pragma unroll
        for (int i = 0; i < 8; ++i) acc[nb][i] = bv;
    }
    {
        const float* arow = &ein[(wv * 16 + lo) * K_EDGE + hi * 2];
#pragma unroll 2
        for (int kb = 0; kb < K_EDGE; kb += 4) {
            v2f a;
            a.x = arow[kb];
            a.y = arow[kb + 1];
            // clamp padded K rows (A is zero there, so product is zero anyway)
            const int k0 = kb + hi * 2;
            const int k0c = (k0     < DIN_E) ? k0     : (DIN_E - 1);
            const int k1c = (k0 + 1 < DIN_E) ? k0 + 1 : (DIN_E - 1);
#pragma unroll
            for (int nb = 0; nb < 4; ++nb) {
                v2f b;
                b.x = We[k0c * D_EDGE + nb * 16 + lo];
                b.y = We[k1c * D_EDGE + nb * 16 + lo];
                acc[nb] = wmma_f32(a, b, acc[nb]);
            }
        }
    }
    // epilogue: feats = e_u + silu(.), duplicated into both halves of e_out
#pragma unroll
    for (int nb = 0; nb < 4; ++nb) {
#pragma unroll
        for (int i = 0; i < 8; ++i) {
            const int m  = i + hi * 8;
            const int ge = eBase + wv * 16 + m;
            if (ge < Eu) {
                const int c = nb * 16 + lo;
                const float eu = ein[(wv * 16 + m) * K_EDGE + c];
                const float f  = eu + silu_f(acc[nb][i]);
                e_out[(size_t)ge * D_EDGE + c]        = f;
                e_out[(size_t)(Eu + ge) * D_EDGE + c] = f;
            }
        }
    }
}

extern "C" void kernel_launch(void* const* d_in, const int* in_sizes, int n_in,
                              void* d_out, int out_size, void* d_ws, size_t ws_size,
                              hipStream_t stream) {
    const float*     s_t       = (const float*)d_in[0];
    const float*     x_t       = (const float*)d_in[1];
    const float*     v_t       = (const float*)d_in[2];
    const float*     e_t       = (const float*)d_in[3];
    const float*     dst_lig_x = (const float*)d_in[4];
    const float*     dst_lig_a = (const float*)d_in[5];
    const float*     dst_lig_e = (const float*)d_in[6];
    const long long* edge_src  = (const long long*)d_in[7];
    const long long* edge_dst  = (const long long*)d_in[8];
    const float*     W1        = (const float*)d_in[9];
    const float*     b1        = (const float*)d_in[10];
    const float*     W2        = (const float*)d_in[11];
    const float*     b2        = (const float*)d_in[12];
    const float*     We        = (const float*)d_in[13];
    const float*     be        = (const float*)d_in[14];

    const int N  = in_sizes[1] / 3;   // x_t is (N,3)
    const int Eu = in_sizes[6] / 5;   // dst_lig_e is (Eu,5)

    float* node_out = (float*)d_out;
    float* v_out    = node_out + (size_t)N * D_NODE;
    float* e_out    = v_out + (size_t)N * 12;

    const int nBlocks = (N + 127) / 128;
    node_mlp_kernel<<<nBlocks, 256, 0, stream>>>(s_t, x_t, v_t, dst_lig_x, dst_lig_a,
                                                 W1, b1, W2, b2, node_out, v_out, N);

    const int eBlocks = (Eu + 127) / 128;
    edge_mlp_kernel<<<eBlocks, 256, 0, stream>>>(e_t, x_t, dst_lig_x, dst_lig_e,
                                                 edge_src, edge_dst, We, be, e_out, Eu);
}